// SequenceTrimmer_798863917405
// MI455X (gfx1250) — compile-verified
//
#include <hip/hip_runtime.h>
#include <stdint.h>

// Problem constants (match reference)
#define B_DIM  256
#define C_DIM  64
#define CV_DIM 4
#define L_DIM  2048
#define NTHR   256   // 8 wave32s per block

// ---------------------------------------------------------------------------
// CDNA5 async global->LDS path (guarded: falls back to plain loads if the
// toolchain lacks the builtin; host pass always takes the fallback).
// Probe feedback: builtin expects int* params (AS1 global, AS3 shared).
// ---------------------------------------------------------------------------
#if defined(__gfx1250__) && __has_builtin(__builtin_amdgcn_global_load_async_to_lds_b32)
#define USE_ASYNC_LDS 1
#else
#define USE_ASYNC_LDS 0
#endif

#if USE_ASYNC_LDS
typedef __attribute__((address_space(1))) int gint_as;
typedef __attribute__((address_space(3))) int lint_as;
__device__ __forceinline__ void async_g2l_b32(const void* g, void* l) {
  __builtin_amdgcn_global_load_async_to_lds_b32((gint_as*)g, (lint_as*)l, 0, 0);
}
__device__ __forceinline__ void wait_async0() {
#if __has_builtin(__builtin_amdgcn_s_wait_asynccnt)
  __builtin_amdgcn_s_wait_asynccnt(0);
#else
  asm volatile("s_wait_asynccnt 0" ::: "memory");
#endif
}
#endif

// Order-preserving encoding: unsigned compare of ou == float compare of f.
__device__ __forceinline__ uint32_t ordered_f32(float f) {
  uint32_t u = __float_as_uint(f);
  return (u & 0x80000000u) ? ~u : (u | 0x80000000u);
}

// ---------------------------------------------------------------------------
// Kernel 1: per-batch-row stable descending argsort of (mask ? -1 : rand),
// plus maxlen = max(popcount(mask), 1).  One block per row.
// LDS: 16KB keys + 8KB rand + 2KB mask stage = 26KB (<< 320KB/WGP).
// ---------------------------------------------------------------------------
__global__ __launch_bounds__(NTHR) void st_sort_kernel(
    const float* __restrict__ rnd, const uint8_t* __restrict__ mask,
    int* __restrict__ perm, int* __restrict__ maxlen) {
  __shared__ unsigned long long keys[L_DIM];
  __shared__ float    s_rand[L_DIM];
  __shared__ uint32_t s_mask[L_DIM / 4];
  __shared__ int      s_cnt;

  const int b   = blockIdx.x;
  const int tid = threadIdx.x;
  if (tid == 0) s_cnt = 0;

  const float*    rrow = rnd + (size_t)b * L_DIM;
  const uint32_t* mrow = (const uint32_t*)(mask + (size_t)b * L_DIM);

#if USE_ASYNC_LDS
  // Stage rand (2048 f32) + mask (512 dwords of packed bytes) via the
  // async path: GLOBAL_LOAD_ASYNC_TO_LDS_B32, tracked by ASYNCcnt.
  #pragma unroll
  for (int k = 0; k < 8; ++k) {
    int i = tid + k * NTHR;
    async_g2l_b32(rrow + i, &s_rand[i]);
  }
  #pragma unroll
  for (int k = 0; k < 2; ++k) {
    int i = tid + k * NTHR;
    async_g2l_b32(mrow + i, &s_mask[i]);
  }
  wait_async0();
  __syncthreads();
#else
  #pragma unroll
  for (int k = 0; k < 8; ++k) { int i = tid + k * NTHR; s_rand[i] = rrow[i]; }
  #pragma unroll
  for (int k = 0; k < 2; ++k) { int i = tid + k * NTHR; s_mask[i] = mrow[i]; }
  __syncthreads();
#endif

  // Build sort keys. Low 11 bits hold (2047 - l) so a descending sort on the
  // composite key reproduces jnp.argsort(-rand_f)'s stable tie-breaking
  // (all masked entries tie at -1.0 and must keep original index order).
  int cnt = 0;
  #pragma unroll
  for (int k = 0; k < 8; ++k) {
    int l = tid + k * NTHR;
    uint32_t mb = (s_mask[l >> 2] >> ((l & 3) * 8)) & 0xFFu;
    float f = mb ? -1.0f : s_rand[l];
    keys[l] = ((unsigned long long)ordered_f32(f) << 11) |
              (unsigned long long)(2047 - l);
    cnt += mb ? 1 : 0;
  }
  atomicAdd(&s_cnt, cnt);
  __syncthreads();

  // In-LDS bitonic sort, descending. 11 stages, 66 steps, 4 active
  // compare-exchanges per thread per step.
  for (int k = 2; k <= L_DIM; k <<= 1) {
    for (int j = k >> 1; j > 0; j >>= 1) {
      #pragma unroll
      for (int t = 0; t < 8; ++t) {
        int i = tid + t * NTHR;
        int ixj = i ^ j;
        if (ixj > i) {
          unsigned long long a = keys[i], c = keys[ixj];
          bool up = ((i & k) == 0);           // descending overall
          if (up ? (a < c) : (a > c)) { keys[i] = c; keys[ixj] = a; }
        }
      }
      __syncthreads();
    }
  }

  if (tid == 0) maxlen[b] = (s_cnt < 1) ? 1 : s_cnt;
  #pragma unroll
  for (int k = 0; k < 8; ++k) {
    int l = tid + k * NTHR;
    perm[(size_t)b * L_DIM + l] = 2047 - (int)(keys[l] & 2047ull);
  }
}

// ---------------------------------------------------------------------------
// Kernel 2: gather along L by perm + zero-trim at maxlen.
// One block per output row: rows 0..B*C-1 -> x, then B*CV rows of v, then B
// rows of mask (emitted as 0.0/1.0 float). perm loads & stores are coalesced;
// the scattered source reads live in an 8KB row (L2/WGP$ resident) and are
// covered with global_prefetch_b8 before the dependent loads.
// ---------------------------------------------------------------------------
__global__ __launch_bounds__(NTHR) void st_gather_kernel(
    const float* __restrict__ x, const float* __restrict__ v,
    const uint8_t* __restrict__ mask, const int* __restrict__ perm,
    const int* __restrict__ maxlen, float* __restrict__ out) {
  const int bid = blockIdx.x;
  const int tid = threadIdx.x;

  const int XROWS = B_DIM * C_DIM;         // 16384
  const int VROWS = B_DIM * CV_DIM;        // 1024

  int b;
  const float*   srcF = nullptr;
  const uint8_t* srcB = nullptr;
  float* dst;
  if (bid < XROWS) {                       // x region
    b    = bid >> 6;
    srcF = x + (size_t)bid * L_DIM;
    dst  = out + (size_t)bid * L_DIM;
  } else if (bid < XROWS + VROWS) {        // v region
    int r = bid - XROWS;
    b    = r >> 2;
    srcF = v + (size_t)r * L_DIM;
    dst  = out + ((size_t)XROWS + r) * L_DIM;
  } else {                                 // mask region
    int r = bid - XROWS - VROWS;
    b    = r;
    srcB = mask + (size_t)r * L_DIM;
    dst  = out + ((size_t)XROWS + VROWS + r) * L_DIM;
  }

  const int  ml   = maxlen[b];
  const int* prow = perm + (size_t)b * L_DIM;

  int p[8];
  #pragma unroll
  for (int k = 0; k < 8; ++k) p[k] = prow[tid + k * NTHR];

  if (srcF) {
    #pragma unroll
    for (int k = 0; k < 8; ++k) __builtin_prefetch(srcF + p[k], 0, 0);
    #pragma unroll
    for (int k = 0; k < 8; ++k) {
      int l = tid + k * NTHR;
      dst[l] = (l < ml) ? srcF[p[k]] : 0.0f;
    }
  } else {
    #pragma unroll
    for (int k = 0; k < 8; ++k) __builtin_prefetch(srcB + p[k], 0, 0);
    #pragma unroll
    for (int k = 0; k < 8; ++k) {
      int l = tid + k * NTHR;
      dst[l] = (l < ml && srcB[p[k]]) ? 1.0f : 0.0f;
    }
  }
}

// ---------------------------------------------------------------------------
// Launch. Workspace layout: int perm[B*L] (2MB) then int maxlen[B].
// Inputs (setup_inputs order): x f32, v f32, mask bool(1B), rand f32.
// Output: concat(x_t, v_t, mask_t) as float.
// ---------------------------------------------------------------------------
extern "C" void kernel_launch(void* const* d_in, const int* in_sizes, int n_in,
                              void* d_out, int out_size, void* d_ws, size_t ws_size,
                              hipStream_t stream) {
  (void)in_sizes; (void)n_in; (void)out_size; (void)ws_size;
  const float*   x    = (const float*)d_in[0];
  const float*   v    = (const float*)d_in[1];
  const uint8_t* mask = (const uint8_t*)d_in[2];
  const float*   rnd  = (const float*)d_in[3];

  int* perm   = (int*)d_ws;
  int* maxlen = perm + (size_t)B_DIM * L_DIM;

  st_sort_kernel<<<B_DIM, NTHR, 0, stream>>>(rnd, mask, perm, maxlen);

  const int nblocks = B_DIM * C_DIM + B_DIM * CV_DIM + B_DIM;  // 17664
  st_gather_kernel<<<nblocks, NTHR, 0, stream>>>(x, v, mask, perm, maxlen,
                                                 (float*)d_out);
}